// Net_58308476010979
// MI455X (gfx1250) — compile-verified
//
#include <hip/hip_runtime.h>
#include <math.h>

typedef __attribute__((ext_vector_type(2))) float v2f;
typedef __attribute__((ext_vector_type(8))) float v8f;

#define NEG_SLOPE 0.2f

// Monotone order-preserving float<->uint encoding for atomicMax-based segment max.
__device__ __forceinline__ unsigned fenc(float f) {
    unsigned u = __float_as_uint(f);
    return u ^ ((unsigned)((int)u >> 31) | 0x80000000u);
}
__device__ __forceinline__ float fdec(unsigned v) {
    unsigned u = v ^ ((unsigned)((int)(~v) >> 31) | 0x80000000u);
    return __uint_as_float(u);
}

__device__ __forceinline__ void edge_sd(const int* __restrict__ EI, int E, int e,
                                        int& s, int& d) {
    if (e < E) { s = EI[e]; d = EI[E + e]; }
    else       { s = e - E; d = s; }          // self-loops appended
}

// ---------------------------------------------------------------------------
// C[M x 64] = A[M x K] * B[K x 64], fp32.
// One wave computes a full 16x64 row-strip: 4 WMMA accumulators share one A
// fragment per K-step (V_WMMA_F32_16X16X4_F32).
// A layout (16x4): lanes 0-15 -> M=lane, v0=K0,v1=K1 ; lanes 16-31 -> K2/K3.
// B layout (4x16): symmetric; C/D: VGPR r -> M=r (lanes<16) / M=r+8.
// ---------------------------------------------------------------------------
template <int K>
__global__ void gemm_wmma(const float* __restrict__ A, const float* __restrict__ B,
                          float* __restrict__ C, int M) {
    int wave = threadIdx.x >> 5;
    int lane = threadIdx.x & 31;
    int mt   = blockIdx.x * (blockDim.x >> 5) + wave;   // one m-tile per wave
    if (mt * 16 >= M) return;                           // wave-uniform
    int half = lane >> 4;                               // 0: K0/K1, 1: K2/K3
    int l15  = lane & 15;
    int row  = mt * 16 + l15;
    if (row >= M) row = M - 1;            // clamp: keep EXEC all-ones for WMMA
    int kofs = half * 2;

    const float* arow = A + (size_t)row * K;
    v8f acc[4] = {};
#pragma unroll 2
    for (int k = 0; k < K; k += 4) {
        v2f a = *(const v2f*)(arow + k + kofs);
        const float* b0 = B + (size_t)(k + kofs) * 64 + l15;   // row k+kofs
        const float* b1 = b0 + 64;                             // row k+kofs+1
#pragma unroll
        for (int nt = 0; nt < 4; ++nt) {
            v2f b; b.x = b0[nt * 16]; b.y = b1[nt * 16];
            acc[nt] = __builtin_amdgcn_wmma_f32_16x16x4_f32(
                          false, a, false, b, (short)0, acc[nt], false, false);
        }
    }

    int rbase = mt * 16 + half * 8;
    if (mt * 16 + 16 <= M) {              // full tile: unguarded straight-line
        float* crow = C + (size_t)rbase * 64 + l15;
#pragma unroll
        for (int r = 0; r < 8; ++r) {
#pragma unroll
            for (int nt = 0; nt < 4; ++nt)
                crow[(size_t)r * 64 + nt * 16] = acc[nt][r];
        }
    } else {                              // boundary tile: per-row guard
#pragma unroll
        for (int r = 0; r < 8; ++r) {
            int rr = rbase + r;
            if (rr < M) {
#pragma unroll
                for (int nt = 0; nt < 4; ++nt)
                    C[(size_t)rr * 64 + nt * 16 + l15] = acc[nt][r];
            }
        }
    }
}

// ---------------------------------------------------------------------------
// Layer 1 (heads=8, ch=8) kernels
// ---------------------------------------------------------------------------
__global__ void attn_prep1(const float* __restrict__ H,
                           const float* __restrict__ asrc, const float* __restrict__ adst,
                           float* __restrict__ AS, float* __restrict__ AD,
                           unsigned* __restrict__ M1, float* __restrict__ S1,
                           float* __restrict__ OUT1, int N) {
    int t = blockIdx.x * blockDim.x + threadIdx.x;
    if (t >= N * 8) return;
    int n = t >> 3, h = t & 7;
    const float* hp = H + (size_t)n * 64 + h * 8;
    float ds = 0.f, dd = 0.f;
#pragma unroll
    for (int c = 0; c < 8; ++c) {
        float v = hp[c];
        ds += v * asrc[h * 8 + c];
        dd += v * adst[h * 8 + c];
    }
    AS[t] = ds; AD[t] = dd; M1[t] = 0u; S1[t] = 0.f;
    float* op = OUT1 + (size_t)n * 64 + h * 8;
#pragma unroll
    for (int c = 0; c < 8; ++c) op[c] = 0.f;
}

__global__ void edge_max1(const int* __restrict__ EI, int E, int Etot,
                          const float* __restrict__ AS, const float* __restrict__ AD,
                          unsigned* __restrict__ M1) {
    int t = blockIdx.x * blockDim.x + threadIdx.x;
    if (t >= Etot * 8) return;
    int e = t >> 3, h = t & 7;
    int s, d; edge_sd(EI, E, e, s, d);
    float v = AS[s * 8 + h] + AD[d * 8 + h];
    v = v > 0.f ? v : NEG_SLOPE * v;
    atomicMax(&M1[d * 8 + h], fenc(v));
}

__global__ void edge_sum1(const int* __restrict__ EI, int E, int Etot,
                          const float* __restrict__ AS, const float* __restrict__ AD,
                          const unsigned* __restrict__ M1, float* __restrict__ S1) {
    int t = blockIdx.x * blockDim.x + threadIdx.x;
    if (t >= Etot * 8) return;
    int e = t >> 3, h = t & 7;
    int s, d; edge_sd(EI, E, e, s, d);
    float v = AS[s * 8 + h] + AD[d * 8 + h];
    v = v > 0.f ? v : NEG_SLOPE * v;
    atomicAdd(&S1[d * 8 + h], __expf(v - fdec(M1[d * 8 + h])));
}

__global__ void edge_scatter1(const int* __restrict__ EI, int E, int Etot,
                              const float* __restrict__ AS, const float* __restrict__ AD,
                              const unsigned* __restrict__ M1, const float* __restrict__ S1,
                              const float* __restrict__ H, float* __restrict__ OUT1) {
    int t = blockIdx.x * blockDim.x + threadIdx.x;
    if (t >= Etot * 8) return;
    int e = t >> 3, h = t & 7;
    int s, d; edge_sd(EI, E, e, s, d);
    float v = AS[s * 8 + h] + AD[d * 8 + h];
    v = v > 0.f ? v : NEG_SLOPE * v;
    float alpha = __expf(v - fdec(M1[d * 8 + h])) / S1[d * 8 + h];
    const float* hp = H + (size_t)s * 64 + h * 8;
    float* op = OUT1 + (size_t)d * 64 + h * 8;
#pragma unroll
    for (int c = 0; c < 8; ++c) atomicAdd(&op[c], hp[c] * alpha);
}

__global__ void elu_k(float* __restrict__ X, int n) {
    int t = blockIdx.x * blockDim.x + threadIdx.x;
    if (t >= n) return;
    float x = X[t];
    X[t] = x > 0.f ? x : expm1f(x);
}

// ---------------------------------------------------------------------------
// Layer 2 (heads=1, ch=64) kernels; OUT == d_out accumulator
// ---------------------------------------------------------------------------
__global__ void attn_prep2(const float* __restrict__ H,
                           const float* __restrict__ asrc, const float* __restrict__ adst,
                           float* __restrict__ AS, float* __restrict__ AD,
                           unsigned* __restrict__ M2, float* __restrict__ S2,
                           float* __restrict__ OUT, int N) {
    int n = blockIdx.x * blockDim.x + threadIdx.x;
    if (n >= N) return;
    const float* hp = H + (size_t)n * 64;
    float ds = 0.f, dd = 0.f;
#pragma unroll 8
    for (int c = 0; c < 64; ++c) {
        float v = hp[c];
        ds += v * asrc[c];
        dd += v * adst[c];
    }
    AS[n] = ds; AD[n] = dd; M2[n] = 0u; S2[n] = 0.f;
    float* op = OUT + (size_t)n * 64;
#pragma unroll 8
    for (int c = 0; c < 64; ++c) op[c] = 0.f;
}

__global__ void edge_max2(const int* __restrict__ EI, int E, int Etot,
                          const float* __restrict__ AS, const float* __restrict__ AD,
                          unsigned* __restrict__ M2) {
    int e = blockIdx.x * blockDim.x + threadIdx.x;
    if (e >= Etot) return;
    int s, d; edge_sd(EI, E, e, s, d);
    float v = AS[s] + AD[d];
    v = v > 0.f ? v : NEG_SLOPE * v;
    atomicMax(&M2[d], fenc(v));
}

__global__ void edge_sum2(const int* __restrict__ EI, int E, int Etot,
                          const float* __restrict__ AS, const float* __restrict__ AD,
                          const unsigned* __restrict__ M2, float* __restrict__ S2) {
    int e = blockIdx.x * blockDim.x + threadIdx.x;
    if (e >= Etot) return;
    int s, d; edge_sd(EI, E, e, s, d);
    float v = AS[s] + AD[d];
    v = v > 0.f ? v : NEG_SLOPE * v;
    atomicAdd(&S2[d], __expf(v - fdec(M2[d])));
}

__global__ void edge_scatter2(const int* __restrict__ EI, int E, int Etot,
                              const float* __restrict__ AS, const float* __restrict__ AD,
                              const unsigned* __restrict__ M2, const float* __restrict__ S2,
                              const float* __restrict__ H, float* __restrict__ OUT) {
    int t = blockIdx.x * blockDim.x + threadIdx.x;
    if (t >= Etot * 8) return;
    int e = t >> 3, g = t & 7;          // 8 channel-groups of 8
    int s, d; edge_sd(EI, E, e, s, d);
    float v = AS[s] + AD[d];
    v = v > 0.f ? v : NEG_SLOPE * v;
    float alpha = __expf(v - fdec(M2[d])) / S2[d];
    const float* hp = H + (size_t)s * 64 + g * 8;
    float* op = OUT + (size_t)d * 64 + g * 8;
#pragma unroll
    for (int c = 0; c < 8; ++c) atomicAdd(&op[c], hp[c] * alpha);
}

// Warp-per-row log_softmax over 64 classes (wave32: 2 elems/lane).
__global__ void log_softmax_k(float* __restrict__ OUT, int N) {
    int warp = threadIdx.x >> 5, lane = threadIdx.x & 31;
    int n = blockIdx.x * (blockDim.x >> 5) + warp;
    if (n >= N) return;
    float* row = OUT + (size_t)n * 64;
    float v0 = row[lane], v1 = row[lane + 32];
    float m = fmaxf(v0, v1);
#pragma unroll
    for (int o = 16; o > 0; o >>= 1) m = fmaxf(m, __shfl_xor(m, o, 32));
    float sum = __expf(v0 - m) + __expf(v1 - m);
#pragma unroll
    for (int o = 16; o > 0; o >>= 1) sum += __shfl_xor(sum, o, 32);
    float lse = m + logf(sum);
    row[lane]      = v0 - lse;
    row[lane + 32] = v1 - lse;
}

// ---------------------------------------------------------------------------
extern "C" void kernel_launch(void* const* d_in, const int* in_sizes, int n_in,
                              void* d_out, int out_size, void* d_ws, size_t ws_size,
                              hipStream_t stream) {
    const float* x   = (const float*)d_in[0];
    const int*   ei  = (const int*)  d_in[1];
    const float* W1  = (const float*)d_in[2];
    const float* a1s = (const float*)d_in[3];
    const float* a1d = (const float*)d_in[4];
    const float* W2  = (const float*)d_in[5];
    const float* a2s = (const float*)d_in[6];
    const float* a2d = (const float*)d_in[7];
    float* out = (float*)d_out;

    const int N    = in_sizes[0] / 256;
    const int E    = in_sizes[1] / 2;
    const int Etot = E + N;

    // Workspace layout (floats): h (N*64) | AS (N*8) | AD (N*8) | M (N*8 u32)
    //                            | S (N*8) | out1 (N*64)   => 160*N floats
    float*    h    = (float*)d_ws;                 // h1, later reused as h2
    float*    AS   = h   + (size_t)N * 64;
    float*    AD   = AS  + (size_t)N * 8;
    unsigned* Mseg = (unsigned*)(AD + (size_t)N * 8);
    float*    Sseg = (float*)Mseg + (size_t)N * 8;
    float*    out1 = Sseg + (size_t)N * 8;

    const int BS = 256;
    int mtiles = (N + 15) / 16;                    // one 16x64 strip per wave
    int gemmB  = (mtiles + (BS / 32) - 1) / (BS / 32);
    int gEH    = (Etot * 8 + BS - 1) / BS;
    int gE     = (Etot + BS - 1) / BS;
    int gNH    = (N * 8 + BS - 1) / BS;
    int gN     = (N + BS - 1) / BS;
    int gNC    = (N * 64 + BS - 1) / BS;
    int gRow   = (N + (BS / 32) - 1) / (BS / 32);

    // ---- layer 1 ----
    gemm_wmma<256><<<gemmB, BS, 0, stream>>>(x, W1, h, N);
    attn_prep1<<<gNH, BS, 0, stream>>>(h, a1s, a1d, AS, AD, Mseg, Sseg, out1, N);
    edge_max1<<<gEH, BS, 0, stream>>>(ei, E, Etot, AS, AD, Mseg);
    edge_sum1<<<gEH, BS, 0, stream>>>(ei, E, Etot, AS, AD, Mseg, Sseg);
    edge_scatter1<<<gEH, BS, 0, stream>>>(ei, E, Etot, AS, AD, Mseg, Sseg, h, out1);
    elu_k<<<gNC, BS, 0, stream>>>(out1, N * 64);

    // ---- layer 2 (h buffer reused for h2; AS/AD/M/S reuse first N entries) ----
    gemm_wmma<64><<<gemmB, BS, 0, stream>>>(out1, W2, h, N);
    attn_prep2<<<gN, BS, 0, stream>>>(h, a2s, a2d, AS, AD, Mseg, Sseg, out, N);
    edge_max2<<<gE, BS, 0, stream>>>(ei, E, Etot, AS, AD, Mseg);
    edge_sum2<<<gE, BS, 0, stream>>>(ei, E, Etot, AS, AD, Mseg, Sseg);
    edge_scatter2<<<gEH, BS, 0, stream>>>(ei, E, Etot, AS, AD, Mseg, Sseg, h, out);

    log_softmax_k<<<gRow, BS, 0, stream>>>(out, N);
}